// RegistrationLoss_43568148251081
// MI455X (gfx1250) — compile-verified
//
#include <hip/hip_runtime.h>
#include <stdint.h>

typedef __attribute__((ext_vector_type(2))) float v2f;
typedef __attribute__((ext_vector_type(8))) float v8f;

// Banded-ones matrix element: B_s[c,c'] = 1 iff 16*s + c - c' in [-4,4]
__device__ __forceinline__ float band(int s, int c, int cp) {
    int d = 16 * s + c - cp;
    return (d >= -4 && d <= 4) ? 1.0f : 0.0f;
}

// ---------------------------------------------------------------------------
// Kernel 0: zero the two scalar accumulators in workspace.
// ---------------------------------------------------------------------------
__global__ void rl_init_kernel(float* ws) {
    if (threadIdx.x < 2) ws[threadIdx.x] = 0.0f;
}

// ---------------------------------------------------------------------------
// Kernel 1: fused 3-D NCC.
// Grid: 512 blocks = 2 batches * 64 (16x16 hw tiles) * 4 z-chunks of 32.
// Block: 256 threads (8 waves). Waves 0..4 own the 5 moment pipelines.
// ---------------------------------------------------------------------------
__global__ __launch_bounds__(256) void rl_ncc_kernel(const float* __restrict__ P,
                                                     const float* __restrict__ T,
                                                     float* ws) {
    __shared__ float pbuf[2][2304];      // 48x48 halo slice of warped, double buffered
    __shared__ float tbuf[2][2304];      // 48x48 halo slice of fixed
    __shared__ float momb[5][2304];      // 5 moment fields for current slice
    __shared__ float Tls[5][3][256];     // stage-A partials (B_a^T * M)
    __shared__ float ring[5][9][256];    // 9-deep z ring of 2D box sums
    __shared__ float red[256];

    const int tid  = threadIdx.x;
    const int wave = tid >> 5;
    const int lane = tid & 31;
    const int hs   = lane >> 4;   // half-wave select (0: lanes 0-15, 1: lanes 16-31)
    const int lr   = lane & 15;

    const int gid  = blockIdx.x;
    const int b    = gid >> 8;
    const int rem  = gid & 255;
    const int zc0  = ((rem >> 6) & 3) * 32;
    const int tile = rem & 63;
    const int h0   = (tile >> 3) * 16;
    const int w0   = (tile & 7) * 16;

    const float* Pb = P + (size_t)b * 2097152u;
    const float* Tb = T + (size_t)b * 2097152u;

    // Issue async global->LDS copies for halo slice si (clamped addresses,
    // exactly 18 wave-level async instructions per slice per wave).
    auto issue = [&](int si) {
        int z = zc0 - 4 + si;
        z = min(max(z, 0), 127);
        float* pd = pbuf[si & 1];
        float* td = tbuf[si & 1];
#pragma unroll
        for (int k = 0; k < 9; ++k) {
            int idx = tid + (k << 8);          // 0..2303
            int r = idx / 48, c = idx - r * 48;
            int h = min(max(h0 - 16 + r, 0), 127);
            int w = min(max(w0 - 16 + c, 0), 127);
            int off = (z << 14) | (h << 7) | w;
            unsigned long long gp = (unsigned long long)(uintptr_t)(Pb + off);
            uint32_t lo = (uint32_t)(uintptr_t)(pd + idx);
            asm volatile("global_load_async_to_lds_b32 %0, %1, off"
                         :: "v"(lo), "v"(gp) : "memory");
            unsigned long long gt = (unsigned long long)(uintptr_t)(Tb + off);
            uint32_t lt = (uint32_t)(uintptr_t)(td + idx);
            asm volatile("global_load_async_to_lds_b32 %0, %1, off"
                         :: "v"(lt), "v"(gt) : "memory");
        }
    };

    issue(0);
    float nccsum = 0.0f;

    for (int zi = 0; zi < 40; ++zi) {
        // Prefetch next slice, then wait until only that batch is outstanding.
        issue(zi + 1);
        asm volatile("s_wait_asynccnt 18" ::: "memory");
        __syncthreads();

        // ---- moment pass: p, t, p^2, t^2, p*t with zero padding ----
        {
            int zz = zc0 - 4 + zi;
            bool zok = (zz >= 0) && (zz < 128);
            const float* ps = pbuf[zi & 1];
            const float* ts = tbuf[zi & 1];
#pragma unroll
            for (int k = 0; k < 9; ++k) {
                int idx = tid + (k << 8);
                int r = idx / 48, c = idx - r * 48;
                int h = h0 - 16 + r, w = w0 - 16 + c;
                bool ok = zok && ((unsigned)h < 128u) && ((unsigned)w < 128u);
                float pv = ok ? ps[idx] : 0.0f;
                float tv = ok ? ts[idx] : 0.0f;
                momb[0][idx] = pv;
                momb[1][idx] = tv;
                momb[2][idx] = pv * pv;
                momb[3][idx] = tv * tv;
                momb[4][idx] = pv * tv;
            }
        }
        __syncthreads();

        // ---- 2D (h,w) 9-wide box sum via banded WMMA: Out = sum_{a,b} B_a^T M_{a,b} B_b
        if (wave < 5) {
            const int m = wave;
            const float* M = momb[m];
            const int phase = zi % 9;

            // stage A: T_b = sum_a B_a^T * M_{a,b}
#pragma unroll
            for (int bc = 0; bc < 3; ++bc) {
                v8f t = {};
#pragma unroll
                for (int a = 0; a < 3; ++a) {
#pragma unroll
                    for (int q = 0; q < 4; ++q) {
                        int kx = 4 * q + 2 * hs;
                        v2f av; av.x = band(a - 1, kx, lr);
                                av.y = band(a - 1, kx + 1, lr);
                        v2f bv; bv.x = M[(a * 16 + kx) * 48 + bc * 16 + lr];
                                bv.y = M[(a * 16 + kx + 1) * 48 + bc * 16 + lr];
                        t = __builtin_amdgcn_wmma_f32_16x16x4_f32(
                                false, av, false, bv, (short)0, t, false, false);
                    }
                }
#pragma unroll
                for (int v = 0; v < 8; ++v)
                    Tls[m][bc][(v + 8 * hs) * 16 + lr] = t[v];
            }

            // stage B: Out = sum_b T_b * B_b
            v8f outv = {};
#pragma unroll
            for (int bc = 0; bc < 3; ++bc) {
#pragma unroll
                for (int q = 0; q < 4; ++q) {
                    int kx = 4 * q + 2 * hs;
                    v2f av; av.x = Tls[m][bc][lr * 16 + kx];
                            av.y = Tls[m][bc][lr * 16 + kx + 1];
                    v2f bv; bv.x = band(bc - 1, kx, lr);
                            bv.y = band(bc - 1, kx + 1, lr);
                    outv = __builtin_amdgcn_wmma_f32_16x16x4_f32(
                               false, av, false, bv, (short)0, outv, false, false);
                }
            }
#pragma unroll
            for (int v = 0; v < 8; ++v)
                ring[m][phase][(v + 8 * hs) * 16 + lr] = outv[v];
        }
        __syncthreads();

        // ---- emit slice zo = zc0 + zi - 8: z box sum + per-voxel NCC ----
        if (zi >= 8) {
            const float inv_n = 1.0f / 729.0f;
            float s[5];
#pragma unroll
            for (int mm = 0; mm < 5; ++mm) {
                float a = 0.0f;
#pragma unroll
                for (int ph = 0; ph < 9; ++ph) a += ring[mm][ph][tid];
                s[mm] = a;
            }
            float ui = s[0] * inv_n, uj = s[1] * inv_n;
            float i2 = s[2] * inv_n - ui * ui;
            float j2 = s[3] * inv_n - uj * uj;
            float ij = s[4] * inv_n - ui * uj;
            nccsum += (ij * ij) / (i2 * j2 + 1e-5f);
        }
    }

    // block reduction + global accumulate
    red[tid] = nccsum;
    __syncthreads();
    for (int off = 128; off > 0; off >>= 1) {
        if (tid < off) red[tid] += red[tid + off];
        __syncthreads();
    }
    if (tid == 0) atomicAdd(&ws[0], red[0]);
}

// ---------------------------------------------------------------------------
// Kernel 2: bending energy (sum of 9 second-difference means, trim-faithful).
// Same-axis pairs: extent 126 along that axis. Cross pairs (i,j): 127 on i,
// 126 on j (trim to 127 then diff).
// ---------------------------------------------------------------------------
__global__ __launch_bounds__(256) void rl_bend_kernel(const float* __restrict__ F,
                                                      float* ws) {
    __shared__ float red[256];
    const int tid = threadIdx.x;
    const int idx = blockIdx.x * 256 + tid;      // < 2*3*128^3

    const int SW = 1, SH = 128, SZ = 16384;
    const int w = idx & 127;
    const int h = (idx >> 7) & 127;
    const int z = (idx >> 14) & 127;

    const float wii = 1.0f / 12386304.0f;  // 2*3*126*128*128
    const float wij = 1.0f / 12289536.0f;  // 2*3*127*126*128

    const float* p = F + idx;
    float f0 = p[0];
    float acc = 0.0f;

    if (z <= 125) { float v = p[2 * SZ] - 2.0f * p[SZ] + f0; acc += wii * v * v; }
    if (h <= 125) { float v = p[2 * SH] - 2.0f * p[SH] + f0; acc += wii * v * v; }
    if (w <= 125) { float v = p[2 * SW] - 2.0f * p[SW] + f0; acc += wii * v * v; }

    {   // (z,h) + (h,z)
        int t = (int)((z <= 126) && (h <= 125)) + (int)((h <= 126) && (z <= 125));
        if (t) { float v = p[SZ + SH] - p[SZ] - p[SH] + f0; acc += (float)t * wij * v * v; }
    }
    {   // (z,w) + (w,z)
        int t = (int)((z <= 126) && (w <= 125)) + (int)((w <= 126) && (z <= 125));
        if (t) { float v = p[SZ + SW] - p[SZ] - p[SW] + f0; acc += (float)t * wij * v * v; }
    }
    {   // (h,w) + (w,h)
        int t = (int)((h <= 126) && (w <= 125)) + (int)((w <= 126) && (h <= 125));
        if (t) { float v = p[SH + SW] - p[SH] - p[SW] + f0; acc += (float)t * wij * v * v; }
    }

    red[tid] = acc;
    __syncthreads();
    for (int off = 128; off > 0; off >>= 1) {
        if (tid < off) red[tid] += red[tid + off];
        __syncthreads();
    }
    if (tid == 0) atomicAdd(&ws[1], red[0]);
}

// ---------------------------------------------------------------------------
// Kernel 3: finalize (total, sim, reg).
// ---------------------------------------------------------------------------
__global__ void rl_fin_kernel(const float* ws, float* out) {
    if (threadIdx.x == 0 && blockIdx.x == 0) {
        float sim = -ws[0] / 4194304.0f;  // mean over 2*1*128^3
        float reg = ws[1];
        out[0] = sim + 0.01f * reg;
        out[1] = sim;
        out[2] = reg;
    }
}

extern "C" void kernel_launch(void* const* d_in, const int* in_sizes, int n_in,
                              void* d_out, int out_size, void* d_ws, size_t ws_size,
                              hipStream_t stream) {
    const float* warped = (const float*)d_in[0];
    const float* fixedv = (const float*)d_in[1];
    const float* flow   = (const float*)d_in[2];
    float* out = (float*)d_out;
    float* ws  = (float*)d_ws;

    rl_init_kernel<<<1, 64, 0, stream>>>(ws);
    rl_ncc_kernel<<<512, 256, 0, stream>>>(warped, fixedv, ws);
    rl_bend_kernel<<<49152, 256, 0, stream>>>(flow, ws);
    rl_fin_kernel<<<1, 1, 0, stream>>>(ws, out);
}